// RFPConv_33552284516501
// MI455X (gfx1250) — compile-verified
//
#include <hip/hip_runtime.h>
#include <hip/hip_bf16.h>

typedef float v2f __attribute__((ext_vector_type(2)));
typedef float v8f __attribute__((ext_vector_type(8)));

#define NN 16384      // nodes
#define NF 128        // features
#define NE 524288     // edges

// ---------------------------------------------------------------- utilities
__global__ void zero_f32(float* p, int n) {
    int i = blockIdx.x * 256 + threadIdx.x;
    if (i < n) p[i] = 0.0f;
}

// ---------------------------------------------------------------- phase 1: degree
__global__ void deg_kernel(const int* __restrict__ ei, float* __restrict__ deg) {
    int e = blockIdx.x * 256 + threadIdx.x;
    if (e < NE) atomicAdd(&deg[ei[2 * e]], 1.0f);
}

// agg init: A[s][f] = 0.5 * deg[s] * x[s][f]   (the "mean" term: deg^2/max(deg,1) == deg)
__global__ void init_agg(const float* __restrict__ x, const float* __restrict__ deg,
                         float* __restrict__ A) {
    int idx = blockIdx.x * 256 + threadIdx.x;   // over NN*NF
    int node = idx >> 7;
    A[idx] = 0.5f * deg[node] * x[idx];
}

// agg scatter: one wave per edge, each lane handles 4 features
__global__ void edge_agg(const int* __restrict__ ei, const float* __restrict__ x,
                         const float* __restrict__ deg, float* __restrict__ A) {
    int gid  = blockIdx.x * 256 + threadIdx.x;
    int e    = gid >> 5;
    int lane = gid & 31;
    if (e >= NE) return;
    int s = ei[2 * e], d = ei[2 * e + 1];
    float w = 0.5f * rsqrtf(deg[s]) * rsqrtf(deg[d]);
    float4 xv = ((const float4*)(x + (size_t)d * NF))[lane];
    float* out = A + (size_t)s * NF + lane * 4;
    atomicAdd(out + 0, w * xv.x);
    atomicAdd(out + 1, w * xv.y);
    atomicAdd(out + 2, w * xv.z);
    atomicAdd(out + 3, w * xv.w);
}

// ---------------------------------------------------------------- phase 2: Householder panel
// Column-j norm^2 over rows i>=j  ->  scal[0]
__global__ void pk_nrm(const float* __restrict__ A, float* __restrict__ scal, int j) {
    __shared__ float red[256];
    int t = threadIdx.x;
    int i = blockIdx.x * 256 + t;
    float v = 0.0f;
    if (i >= j && i < NN) { float a = A[(size_t)i * NF + j]; v = a * a; }
    red[t] = v;
    __syncthreads();
    for (int s = 128; s > 0; s >>= 1) {
        if (t < s) red[t] += red[t + s];
        __syncthreads();
    }
    if (t == 0) atomicAdd(&scal[0], red[0]);
}

// Build reflector in place: column j becomes v (0 above diag, 1 on diag, scaled below).
// Also writes tau[j], zeroes wvec (for pk_dot) and scal[0] (for next pk_nrm).
__global__ void pk_house(float* __restrict__ A, float* __restrict__ scal,
                         float* __restrict__ tauv, float* __restrict__ wvec, int j) {
    __shared__ float sh_scale;
    int t = threadIdx.x;   // 256 threads
    if (t == 0) {
        float nrm2  = scal[0];
        float alpha = A[(size_t)j * NF + j];
        float tau, scale;
        if (nrm2 > 0.0f) {
            float nrm  = sqrtf(nrm2);
            float beta = (alpha >= 0.0f) ? -nrm : nrm;
            tau   = (beta - alpha) / beta;
            scale = 1.0f / (alpha - beta);
        } else { tau = 0.0f; scale = 0.0f; }
        tauv[j]  = tau;
        sh_scale = scale;
        scal[0]  = 0.0f;
    }
    if (t < NF) wvec[t] = 0.0f;
    __syncthreads();
    float scale = sh_scale;
    for (int i = t; i < NN; i += 256) {
        float v;
        if (i < j)       v = 0.0f;
        else if (i == j) v = 1.0f;
        else             v = A[(size_t)i * NF + j] * scale;
        A[(size_t)i * NF + j] = v;
    }
}

// wvec[c] = v^T A[:,c]  (v = column j; zero above diag so full row range is safe)
__global__ void pk_dot(const float* __restrict__ A, float* __restrict__ wvec, int j) {
    int c  = threadIdx.x;        // 0..127
    int i0 = blockIdx.x * 128;   // 128 row-blocks
    float acc = 0.0f;
    for (int r = 0; r < 128; ++r) {
        size_t row = (size_t)(i0 + r) * NF;
        float vj = A[row + j];
        acc += vj * A[row + c];
    }
    if (c > j) atomicAdd(&wvec[c], acc);
}

// A[i][c] -= tau * v[i] * wvec[c]  for c>j (v[i]==0 for i<j -> skip store)
__global__ void pk_upd(const float* __restrict__ tauv, const float* __restrict__ wvec,
                       float* __restrict__ A, int j) {
    int c = threadIdx.x;         // 0..127
    int i = blockIdx.x;          // 0..NN-1
    float vj = A[(size_t)i * NF + j];
    if (vj != 0.0f && c > j)
        A[(size_t)i * NF + c] -= tauv[j] * vj * wvec[c];
}

// ---------------------------------------------------------------- phase 3: compact WY
// S = V^T V (only strictly-upper part is consumed)
__global__ void s_kernel(const float* __restrict__ A, float* __restrict__ S) {
    int r  = blockIdx.x * 16 + threadIdx.x;
    int c  = blockIdx.y * 16 + threadIdx.y;
    int i0 = blockIdx.z * 256;
    float acc = 0.0f;
    for (int k = 0; k < 256; ++k) {
        size_t row = (size_t)(i0 + k) * NF;
        acc += A[row + r] * A[row + c];
    }
    atomicAdd(&S[r * NF + c], acc);
}

// T: upper-triangular WY factor. T[:j,j] = -tau_j * T[:j,:j] * S[:j,j]; T[j,j]=tau_j
__global__ void t_kernel(const float* __restrict__ S, const float* __restrict__ tauv,
                         float* __restrict__ T) {
    int r = threadIdx.x;   // 128 threads
    for (int c = 0; c < NF; ++c) T[r * NF + c] = 0.0f;
    __syncthreads();
    for (int j = 0; j < NF; ++j) {
        float tj = tauv[j];
        if (r < j) {
            float z = 0.0f;
            for (int k = r; k < j; ++k) z += T[r * NF + k] * S[k * NF + j];
            T[r * NF + j] = -tj * z;
        } else if (r == j) {
            T[j * NF + j] = tj;
        }
        __syncthreads();
    }
}

// W = V * T  (T upper-triangular)
__global__ void w_kernel(const float* __restrict__ A, const float* __restrict__ T,
                         float* __restrict__ W) {
    __shared__ float vrow[NF];
    int i = blockIdx.x;
    int c = threadIdx.x;
    vrow[c] = A[(size_t)i * NF + c];
    __syncthreads();
    float acc = 0.0f;
    for (int k = 0; k <= c; ++k) acc += vrow[k] * T[k * NF + c];
    W[(size_t)i * NF + c] = acc;
}

// ---------------------------------------------------------------- phase 4: Q = I - W V^T
// 64x64 macro-tile / block (4 waves), K=128 via two 64-wide LDS chunks,
// each wave: 16x64 strip = 4 x v_wmma_f32_16x16x4_f32 accumulators.
__global__ __launch_bounds__(128) void q_kernel(const float* __restrict__ W,
                                                const float* __restrict__ V,
                                                float* __restrict__ Q) {
    __shared__ float sW[64 * 64];
    __shared__ float sV[64 * 64];
    const int m0 = blockIdx.x * 64;
    const int n0 = blockIdx.y * 64;

    const int wave = threadIdx.x >> 5;
    const int lane = threadIdx.x & 31;
    const int half = lane >> 4;      // ISA 7.12.2: lanes 0-15 hold K={0,1}, 16-31 K={2,3}
    const int l16  = lane & 15;
    const int mrow = wave * 16 + l16;

    v8f acc0 = {}, acc1 = {}, acc2 = {}, acc3 = {};

    const float4* gW = (const float4*)(W + (size_t)m0 * NF);
    const float4* gV = (const float4*)(V + (size_t)n0 * NF);
    float4* lW = (float4*)sW;
    float4* lV = (float4*)sV;

    for (int kc = 0; kc < 2; ++kc) {
        // stage 64 rows x 64 K-values of W and V
        for (int t = threadIdx.x; t < 1024; t += 128) {   // 1024 float4 per array
            int r  = t >> 4;
            int k4 = t & 15;
            lW[t] = gW[(size_t)r * 32 + kc * 16 + k4];
            lV[t] = gV[(size_t)r * 32 + kc * 16 + k4];
        }
        __syncthreads();

        for (int kk = 0; kk < 16; ++kk) {
            int kb = kk * 4 + half * 2;
            v2f a;
            a.x = sW[mrow * 64 + kb];
            a.y = sW[mrow * 64 + kb + 1];
            v2f b0, b1, b2, b3;
            b0.x = sV[(0 * 16 + l16) * 64 + kb]; b0.y = sV[(0 * 16 + l16) * 64 + kb + 1];
            b1.x = sV[(1 * 16 + l16) * 64 + kb]; b1.y = sV[(1 * 16 + l16) * 64 + kb + 1];
            b2.x = sV[(2 * 16 + l16) * 64 + kb]; b2.y = sV[(2 * 16 + l16) * 64 + kb + 1];
            b3.x = sV[(3 * 16 + l16) * 64 + kb]; b3.y = sV[(3 * 16 + l16) * 64 + kb + 1];
            acc0 = __builtin_amdgcn_wmma_f32_16x16x4_f32(false, a, false, b0, (short)0, acc0, false, false);
            acc1 = __builtin_amdgcn_wmma_f32_16x16x4_f32(false, a, false, b1, (short)0, acc1, false, false);
            acc2 = __builtin_amdgcn_wmma_f32_16x16x4_f32(false, a, false, b2, (short)0, acc2, false, false);
            acc3 = __builtin_amdgcn_wmma_f32_16x16x4_f32(false, a, false, b3, (short)0, acc3, false, false);
        }
        __syncthreads();
    }

    // C/D layout (ISA 7.12.2): VGPR i -> row M=i (lanes 0-15) / M=i+8 (lanes 16-31), N=lane%16
    v8f accs[4] = {acc0, acc1, acc2, acc3};
    for (int nt = 0; nt < 4; ++nt) {
        int gn = n0 + nt * 16 + l16;
        for (int i = 0; i < 8; ++i) {
            int gm = m0 + wave * 16 + i + half * 8;
            float val = -accs[nt][i];
            if (gm == gn) val += 1.0f;
            Q[(size_t)gm * NN + gn] = val;
        }
    }
}

// ---------------------------------------------------------------- host
extern "C" void kernel_launch(void* const* d_in, const int* in_sizes, int n_in,
                              void* d_out, int out_size, void* d_ws, size_t ws_size,
                              hipStream_t stream) {
    const float* x  = (const float*)d_in[0];   // (16384,128) f32
    const int*   ei = (const int*)d_in[1];     // (524288,2) i32
    float* Q = (float*)d_out;                  // (16384,16384) f32

    char* ws = (char*)d_ws;
    float* deg  = (float*)ws;  ws += (size_t)NN * 4;                 // 64 KB
    float* A    = (float*)ws;  ws += (size_t)NN * NF * 4;            // 8 MB (panel -> V)
    float* W    = (float*)ws;  ws += (size_t)NN * NF * 4;            // 8 MB
    float* S    = (float*)ws;  ws += (size_t)NF * NF * 4;            // 64 KB
    float* T    = (float*)ws;  ws += (size_t)NF * NF * 4;            // 64 KB
    float* tauv = (float*)ws;  ws += 512;
    float* wvec = (float*)ws;  ws += 512;
    float* scal = (float*)ws;  ws += 256;

    // init
    zero_f32<<<(NN + 255) / 256, 256, 0, stream>>>(deg, NN);
    zero_f32<<<(NF * NF + 255) / 256, 256, 0, stream>>>(S, NF * NF);
    zero_f32<<<1, 256, 0, stream>>>(scal, 8);

    // phase 1: aggregation
    deg_kernel<<<(NE + 255) / 256, 256, 0, stream>>>(ei, deg);
    init_agg<<<(NN * NF) / 256, 256, 0, stream>>>(x, deg, A);
    edge_agg<<<NE / 8, 256, 0, stream>>>(ei, x, deg, A);   // 8 waves (edges) / block

    // phase 2: Householder panel factorization (V stored in A, unit diag, zero above)
    for (int j = 0; j < NF; ++j) {
        pk_nrm<<<64, 256, 0, stream>>>(A, scal, j);
        pk_house<<<1, 256, 0, stream>>>(A, scal, tauv, wvec, j);
        if (j < NF - 1) {
            pk_dot<<<128, 128, 0, stream>>>(A, wvec, j);
            pk_upd<<<NN, 128, 0, stream>>>(tauv, wvec, A, j);
        }
    }

    // phase 3: compact WY
    s_kernel<<<dim3(8, 8, 64), dim3(16, 16), 0, stream>>>(A, S);
    t_kernel<<<1, 128, 0, stream>>>(S, tauv, T);
    w_kernel<<<NN, 128, 0, stream>>>(A, T, W);

    // phase 4: Q = I - W V^T  (rank-128 update via f32 WMMA)
    q_kernel<<<dim3(NN / 64, NN / 64), 128, 0, stream>>>(W, A, Q);
}